// SpatialProbLoss_63986422776311
// MI455X (gfx1250) — compile-verified
//
#include <hip/hip_runtime.h>
#include <math.h>

typedef __attribute__((ext_vector_type(2))) float v2f;
typedef __attribute__((ext_vector_type(8))) float v8f;

#define N_PRED 8192
#define N_TRUE 4096
#define TILES_N (N_PRED / 16)   // 512
#define TILES_M (N_TRUE / 16)   // 256
#define WAVES_PER_BLOCK 8
#define P2_THREADS 256
#define INF_F __builtin_inff()

// ---------------------------------------------------------------------------
// Phase 1: distance matrix via V_WMMA_F32_16X16X4_F32 (K=4 exactly fits xyz+1).
//   d2[m][n] = (|t_m|^2 - 2 t_m . p_n) + |p_n|^2
//   A row m  = (tx, ty, tz, |t_m|^2)         (16x4, K=4)
//   B col n  = (-2px, -2py, -2pz, 1)         (4x16)
//   C        = 0
// The leftover |p_n|^2 is column-only, and lane L's output column (N = L&15)
// is exactly the pred row lane L loaded -> epilogue adds the lane's own psq.
// No cross-lane traffic at all; one wave computes one 16x16 tile; EXEC all-1s.
// ---------------------------------------------------------------------------
__global__ __launch_bounds__(256) void spl_dist_wmma_kernel(
    const float4* __restrict__ pred, const float4* __restrict__ truec,
    float* __restrict__ dist)
{
    const int lane = threadIdx.x & 31;
    const int wave = threadIdx.x >> 5;
    const int tile = blockIdx.x * WAVES_PER_BLOCK + wave;
    const int tm   = tile / TILES_N;     // true-tile (M)
    const int tn   = tile % TILES_N;     // pred-tile (N)
    const int r16  = lane & 15;
    const int half = lane >> 4;          // 0: lanes 0-15, 1: lanes 16-31

    // A operand: lane holds row M = r16; lanes 0-15 -> (K0,K1), 16-31 -> (K2,K3)
    const float4 t   = truec[tm * 16 + r16];
    const float  tsq = t.x * t.x + t.y * t.y + t.z * t.z;
    v2f a;
    a.x = half ? t.z : t.x;
    a.y = half ? tsq : t.y;

    // B operand: lane holds col N = r16; VGPR0 = K-row (0|2), VGPR1 = K-row (1|3)
    const float4 p   = pred[tn * 16 + r16];
    const float  psq = p.x * p.x + p.y * p.y + p.z * p.z;
    v2f b;
    b.x = half ? (-2.0f * p.z) : (-2.0f * p.x);
    b.y = half ? 1.0f          : (-2.0f * p.y);

    v8f c = {};   // C = 0

    // |t|^2 - 2 t.p   (v_wmma_f32_16x16x4_f32)
    v8f d = __builtin_amdgcn_wmma_f32_16x16x4_f32(
        /*neg_a=*/false, a, /*neg_b=*/false, b,
        /*c_mod=*/(short)0, c, /*reuse_a=*/false, /*reuse_b=*/false);

    // Epilogue: + psq (lane's own column), mask on d2 (monotonic with dist),
    // single-instruction v_sqrt_f32 for the stored value.
    const bool colvalid = (p.w >= 0.5f);
    float* out_base = dist + (size_t)(tm * 16) * N_PRED + (size_t)tn * 16 + r16;
#pragma unroll
    for (int v = 0; v < 8; ++v) {
        float d2 = d[v] + psq;
        d2 = d2 < 0.0f ? 0.0f : d2;            // guard tiny negatives
        const bool ok = colvalid && (d2 <= 1.0f);
        float r = ok ? __builtin_amdgcn_sqrtf(d2) : INF_F;
        out_base[(size_t)(v + 8 * half) * N_PRED] = r;
    }
}

// ---------------------------------------------------------------------------
// Phase 2: serial greedy matching (one workgroup; chain is inherently serial).
// Suppression bitmap + probs staged in LDS; argmin = per-thread strided min,
// wave32 __shfl_down reduce, cross-wave LDS reduce, thread-0 serial update.
// Next row is prefetched (global_prefetch_b8) during the current reduction.
// ---------------------------------------------------------------------------
__global__ __launch_bounds__(P2_THREADS) void spl_greedy_kernel(
    const float* __restrict__ dist, const float4* __restrict__ pred,
    const float4* __restrict__ truec, float* __restrict__ out)
{
    __shared__ unsigned char suppressed[N_PRED];
    __shared__ float pp[N_PRED];
    __shared__ float tp[N_TRUE];
    __shared__ float wmin[P2_THREADS / 32];
    __shared__ int   widx[P2_THREADS / 32];
    __shared__ int   num_true_s;

    const int tid  = threadIdx.x;
    const int lane = tid & 31;
    const int wv   = tid >> 5;

    if (tid == 0) num_true_s = 0;
    __syncthreads();

    for (int j = tid; j < N_PRED; j += P2_THREADS) {
        suppressed[j] = 0;
        pp[j] = pred[j].w;
    }
    int local_true = 0;
    for (int i = tid; i < N_TRUE; i += P2_THREADS) {
        float v = truec[i].w;
        tp[i] = v;
        local_true += (v >= 0.5f) ? 1 : 0;
    }
    atomicAdd(&num_true_s, local_true);
    __syncthreads();

    float sum_dist = 0.0f, sum_sq = 0.0f;
    int   cnt = 0;   // only thread 0's copies are meaningful

    for (int row = 0; row < N_TRUE; ++row) {
        const float* drow = dist + (size_t)row * N_PRED;

        float best = INF_F;
        int   bidx = 0;
        for (int j = tid; j < N_PRED; j += P2_THREADS) {
            float v = suppressed[j] ? INF_F : drow[j];
            if (v < best) { best = v; bidx = j; }
        }

        // Hide next row's latency behind the reduction/serial section.
        if (row + 1 < N_TRUE)
            __builtin_prefetch(drow + N_PRED + tid * (N_PRED / P2_THREADS), 0, 0);

        // wave32 argmin reduce (lowest index wins ties, matching argmin-first)
        for (int off = 16; off > 0; off >>= 1) {
            float ov = __shfl_down(best, off, 32);
            int   oi = __shfl_down(bidx, off, 32);
            if (ov < best || (ov == best && oi < bidx)) { best = ov; bidx = oi; }
        }
        if (lane == 0) { wmin[wv] = best; widx[wv] = bidx; }
        __syncthreads();

        if (tid == 0) {
            float fb = wmin[0];
            int   fi = widx[0];
            for (int w = 1; w < P2_THREADS / 32; ++w) {
                float ov = wmin[w]; int oi = widx[w];
                if (ov < fb || (ov == fb && oi < fi)) { fb = ov; fi = oi; }
            }
            const float tprob   = tp[row];
            const bool  matched = (tprob >= 0.5f) && (fb < INF_F);
            if (matched) {
                suppressed[fi] = 1;
                sum_dist += fb;
                float dp = tprob - pp[fi];
                sum_sq  += dp * dp;
                cnt     += 1;
            }
        }
        __syncthreads();   // publish suppression before the next row
    }

    if (tid == 0) {
        const float num_true  = (float)num_true_s;
        const float cnt_f     = (float)cnt;
        const float unmatched = num_true - cnt_f;
        const float denom     = fmaxf(cnt_f, 1.0f);
        const float spatial   = 10.0f * unmatched + (cnt > 0 ? sum_dist / denom : 0.0f);
        const float prob      = unmatched + (cnt > 0 ? sum_sq / denom : 0.0f);
        out[0] = spatial + prob;
    }
}

extern "C" void kernel_launch(void* const* d_in, const int* in_sizes, int n_in,
                              void* d_out, int out_size, void* d_ws, size_t ws_size,
                              hipStream_t stream) {
    (void)in_sizes; (void)n_in; (void)out_size; (void)ws_size;
    const float4* pred  = (const float4*)d_in[0];   // (8192, 4)
    const float4* truec = (const float4*)d_in[1];   // (4096, 4)
    float* dist = (float*)d_ws;                     // 4096*8192*4 = 128 MiB scratch
    float* out  = (float*)d_out;

    const int total_tiles = TILES_M * TILES_N;      // 131072 waves
    spl_dist_wmma_kernel<<<total_tiles / WAVES_PER_BLOCK, 256, 0, stream>>>(pred, truec, dist);
    spl_greedy_kernel<<<1, P2_THREADS, 0, stream>>>(dist, pred, truec, out);
}